// HybridMemory_21277267984510
// MI455X (gfx1250) — compile-verified
//
#include <hip/hip_runtime.h>

#define NF 256       // feature dim (K)
#define NS 65536     // memory-bank samples (N)
#define NC 2000      // classes
#define NCP 2048     // classes padded to multiple of 128
#define BATCH 256
#define TEMP 0.05f
#define INS_TEMP 0.09f

typedef __attribute__((ext_vector_type(2))) float v2f;
typedef __attribute__((ext_vector_type(8))) float v8f;

// ---------------- CDNA5 async global->LDS copy (bypasses VGPRs, ASYNCcnt-tracked) ----------
__device__ __forceinline__ void async_b128(const float* g, float* lds_generic) {
  // LDS byte offset == low 32 bits of the flat shared-aperture address (ISA 10.2).
  unsigned int l = (unsigned int)(size_t)lds_generic;
  unsigned long long ga = (unsigned long long)(size_t)g;
  asm volatile("global_load_async_to_lds_b128 %0, %1, off"
               :
               : "v"(l), "v"(ga)
               : "memory");
}
__device__ __forceinline__ void wait_async_le8() {
  asm volatile("s_wait_asynccnt 8" ::: "memory");
}
__device__ __forceinline__ void wait_async_0() {
  asm volatile("s_wait_asynccnt 0" ::: "memory");
}

// Issue one K-chunk (128 rows x 64 floats) of B into an LDS buffer: 8 async b128 per thread,
// i.e. exactly 8 async instructions per wave.
__device__ __forceinline__ void issue_B_chunk(const float* __restrict__ Bglob, int f0,
                                              int kc, float* Bs, int tid) {
#pragma unroll
  for (int i = 0; i < 8; ++i) {
    int idx = tid + i * 256;
    int r = idx >> 4, c = (idx & 15) << 2;
    async_b128(Bglob + (size_t)(f0 + r) * NF + kc + c, &Bs[r * 68 + c]);
  }
}

// ---------------- reductions ----------------
__device__ __forceinline__ float blockReduceSum(float v, float* sm) {
#pragma unroll
  for (int o = 16; o > 0; o >>= 1) v += __shfl_down(v, o, 32);
  int lane = threadIdx.x & 31, w = threadIdx.x >> 5;
  if (lane == 0) sm[w] = v;
  __syncthreads();
  int nw = (blockDim.x + 31) >> 5;
  v = (threadIdx.x < nw) ? sm[threadIdx.x] : 0.0f;
  if (w == 0) {
#pragma unroll
    for (int o = 4; o > 0; o >>= 1) v += __shfl_down(v, o, 32);
  }
  __syncthreads();
  return v;  // valid on thread 0
}

// ---------------- init ----------------
__global__ void k_zero(float* p, int n) {
  int i = blockIdx.x * blockDim.x + threadIdx.x;
  if (i < n) p[i] = 0.0f;
}

__global__ void k_gather(const int* labels, const int* labels2, const int* indexes,
                         int* tg, int* tg2) {
  int i = threadIdx.x;
  if (i < BATCH) {
    int idx = indexes[i];
    tg[i] = labels[idx];
    tg2[i] = labels2[idx];
  }
}

// ---------------- class aggregation: F_cls[c] = sum_{labels[j]==c} features[j] ----------------
// 250 blocks, 8 classes each, privatized in LDS -> zero global atomics.
__global__ __launch_bounds__(256) void k_agg(const float* __restrict__ features,
                                             const int* __restrict__ labels,
                                             float* __restrict__ Fcls,
                                             float* __restrict__ nums) {
  __shared__ float acc[8 * 256];
  __shared__ int qj[256];
  __shared__ int qc[256];
  __shared__ int qcnt;
  __shared__ int cnt8[8];
  const int tid = threadIdx.x;
  const int c0 = blockIdx.x * 8;
#pragma unroll
  for (int i = 0; i < 8; ++i) acc[i * 256 + tid] = 0.0f;
  if (tid < 8) cnt8[tid] = 0;
  for (int base = 0; base < NS; base += 256) {
    if (tid == 0) qcnt = 0;
    __syncthreads();
    int c = labels[base + tid] - c0;
    if ((unsigned)c < 8u) {
      int p = atomicAdd(&qcnt, 1);
      qj[p] = base + tid;
      qc[p] = c;
      atomicAdd(&cnt8[c], 1);
    }
    __syncthreads();
    int n = qcnt;
    for (int q = 0; q < n; ++q)
      acc[qc[q] * 256 + tid] += features[(size_t)qj[q] * NF + tid];
    __syncthreads();
  }
#pragma unroll
  for (int i = 0; i < 8; ++i) Fcls[(size_t)(c0 + i) * NF + tid] = acc[i * 256 + tid];
  if (tid < 8) nums[c0 + tid] = (float)cnt8[tid];
}

// ---------------- WMMA GEMM #1: sim = inputs @ Fcls.T / TEMP / nums ----------------
// grid (NCP/128, BATCH/16), 256 threads (8 waves; one 16x16 C-tile per wave).
// Double-buffered async global->LDS staging of B.
__global__ __launch_bounds__(256) void k_gemm_sim(const float* __restrict__ inputs,
                                                  const float* __restrict__ Fcls,
                                                  const float* __restrict__ nums,
                                                  float* __restrict__ sim) {
  __shared__ float As[16 * 260];   // padded: bank-conflict-free fragment reads
  __shared__ float Bs0[128 * 68];
  __shared__ float Bs1[128 * 68];
  const int tid = threadIdx.x;
  const int m0 = blockIdx.y * 16;
  const int c0 = blockIdx.x * 128;

  issue_B_chunk(Fcls, c0, 0, Bs0, tid);  // prefetch chunk 0

#pragma unroll
  for (int i = 0; i < 4; ++i) {
    int idx = tid + i * 256;
    int r = idx >> 6, c = (idx & 63) << 2;
    *(float4*)&As[r * 260 + c] = *(const float4*)&inputs[(size_t)(m0 + r) * NF + c];
  }
  const int lane = tid & 31, w = tid >> 5;
  const int li = lane & 15, hi = lane >> 4;
  v8f accv = {};
#pragma unroll
  for (int ck = 0; ck < 4; ++ck) {
    if (ck < 3) {
      issue_B_chunk(Fcls, c0, (ck + 1) * 64, (ck & 1) ? Bs0 : Bs1, tid);
      wait_async_le8();   // current buffer's 8 fills done; next 8 stay in flight
    } else {
      wait_async_0();
    }
    __syncthreads();
    const float* Bcur = (ck & 1) ? Bs1 : Bs0;
    const int kc = ck * 64;
#pragma unroll
    for (int kk = 0; kk < 64; kk += 4) {
      v2f a = *(const v2f*)&As[li * 260 + kc + kk + 2 * hi];
      v2f b = *(const v2f*)&Bcur[(w * 16 + li) * 68 + kk + 2 * hi];
      accv = __builtin_amdgcn_wmma_f32_16x16x4_f32(false, a, false, b, (short)0, accv,
                                                   false, false);
    }
    __syncthreads();  // all reads done before next iteration overwrites the other buffer
  }
  int c = c0 + w * 16 + li;
  float nv = nums[c];
  float s = (1.0f / TEMP) / (nv > 0.0f ? nv : 1.0f);
#pragma unroll
  for (int v = 0; v < 8; ++v)
    sim[(size_t)(m0 + v + 8 * hi) * NCP + c] = accv[v] * s;
}

// ---------------- WMMA GEMM #2: outputs = inputs @ features.T, fused ins-loss masked sums ----
// grid (NS/128, BATCH/16), 256 threads. Same async double-buffered staging.
__global__ __launch_bounds__(256) void k_gemm_ins(const float* __restrict__ inputs,
                                                  const float* __restrict__ features,
                                                  const int* __restrict__ labels,
                                                  const int* __restrict__ labels2,
                                                  const int* __restrict__ tg,
                                                  const int* __restrict__ tg2,
                                                  float* __restrict__ pos_e,
                                                  float* __restrict__ neg_e,
                                                  int* __restrict__ n_pos,
                                                  int* __restrict__ n_neg) {
  __shared__ float As[16 * 260];
  __shared__ float Bs0[128 * 68];
  __shared__ float Bs1[128 * 68];
  __shared__ int tT[16], tT2[16];
  const int tid = threadIdx.x;
  const int m0 = blockIdx.y * 16;
  const int f0 = blockIdx.x * 128;

  issue_B_chunk(features, f0, 0, Bs0, tid);  // prefetch chunk 0

#pragma unroll
  for (int i = 0; i < 4; ++i) {
    int idx = tid + i * 256;
    int r = idx >> 6, c = (idx & 63) << 2;
    *(float4*)&As[r * 260 + c] = *(const float4*)&inputs[(size_t)(m0 + r) * NF + c];
  }
  if (tid < 16) { tT[tid] = tg[m0 + tid]; tT2[tid] = tg2[m0 + tid]; }
  const int lane = tid & 31, w = tid >> 5;
  const int li = lane & 15, hi = lane >> 4;
  v8f accv = {};
#pragma unroll
  for (int ck = 0; ck < 4; ++ck) {
    if (ck < 3) {
      issue_B_chunk(features, f0, (ck + 1) * 64, (ck & 1) ? Bs0 : Bs1, tid);
      wait_async_le8();
    } else {
      wait_async_0();
    }
    __syncthreads();
    const float* Bcur = (ck & 1) ? Bs1 : Bs0;
    const int kc = ck * 64;
#pragma unroll
    for (int kk = 0; kk < 64; kk += 4) {
      v2f a = *(const v2f*)&As[li * 260 + kc + kk + 2 * hi];
      v2f b = *(const v2f*)&Bcur[(w * 16 + li) * 68 + kk + 2 * hi];
      accv = __builtin_amdgcn_wmma_f32_16x16x4_f32(false, a, false, b, (short)0, accv,
                                                   false, false);
    }
    __syncthreads();
  }
  // epilogue: instances = outputs / 0.09; masks are rare -> conditional atomics only.
  int j = f0 + w * 16 + li;
  int la = labels[j], la2 = labels2[j];
#pragma unroll
  for (int v = 0; v < 8; ++v) {
    int ml = v + 8 * hi;
    bool inA = (la == tT[ml]);
    bool inB = (la2 == tT2[ml]);
    if (inA != inB) {
      float e = __expf(accv[v] * (1.0f / INS_TEMP));
      int mi = m0 + ml;
      if (inA) { atomicAdd(&pos_e[mi], e); atomicAdd(&n_pos[mi], 1); }
      else     { atomicAdd(&neg_e[mi], e); atomicAdd(&n_neg[mi], 1); }
    }
  }
}

// ---------------- focal loss: one block per sample ----------------
__global__ __launch_bounds__(256) void k_focal(const float* __restrict__ sim,
                                               const float* __restrict__ nums,
                                               const int* __restrict__ tg,
                                               float* __restrict__ focal_sum) {
  __shared__ float sm[8];
  int b = blockIdx.x;
  float s = 0.0f;
  for (int c = threadIdx.x; c < NC; c += blockDim.x)
    if (nums[c] > 0.0f) s += __expf(sim[(size_t)b * NCP + c]);
  s = blockReduceSum(s, sm);
  if (threadIdx.x == 0) {
    int t = tg[b];
    float et = (nums[t] > 0.0f) ? __expf(sim[(size_t)b * NCP + t]) : 0.0f;
    float p = et / (s + 1e-6f);
    float lb = -(1.0f - p) * (1.0f - p) * __logf(p + 1e-6f);
    atomicAdd(focal_sum, lb);
  }
}

// ---------------- contrastive loss ----------------
__global__ __launch_bounds__(256) void k_contras(const float* __restrict__ a,
                                                 const float* __restrict__ bm,
                                                 float* __restrict__ csum) {
  __shared__ float sm[8];
  int i = blockIdx.x;
  float av = a[(size_t)i * NF + threadIdx.x];
  float bv = bm[(size_t)i * NF + threadIdx.x];
  float d  = blockReduceSum(av * bv, sm);
  float na = blockReduceSum(av * av, sm);
  float nb = blockReduceSum(bv * bv, sm);
  if (threadIdx.x == 0) atomicAdd(csum, -d * rsqrtf(na) * rsqrtf(nb));
}

// ---------------- final combine ----------------
__global__ __launch_bounds__(256) void k_final(const float* pos_e, const float* neg_e,
                                               const int* n_pos, const int* n_neg,
                                               const float* focal_sum, const float* csum,
                                               const int* epoch, const int* back,
                                               float* out) {
  __shared__ float sm[8];
  int i = threadIdx.x;
  float per = 0.0f;
  int np = n_pos[i], nn = n_neg[i];
  if (np > 0 && nn > 0) {
    float pe = pos_e[i], ne = neg_e[i];
    float ins = pe / (pe + ne + 1e-6f);
    per = -__logf(ins + 1e-6f) / (float)(np > 1 ? np : 1);
  }
  float isum = blockReduceSum(per, sm);
  if (i == 0) {
    float focal   = focal_sum[0] * (1.0f / BATCH);
    float contras = csum[0] * (1.0f / BATCH);
    float insm    = isum * (1.0f / BATCH);
    int bk = back[0], ep = epoch[0];
    float r;
    if (bk == 1) { r = focal + 0.25f * contras; if (ep >= 30) r += 0.2f * insm; }
    else if (bk == 2) r = focal + 0.25f * contras;
    else r = focal;
    out[0] = r;
  }
}

extern "C" void kernel_launch(void* const* d_in, const int* in_sizes, int n_in,
                              void* d_out, int out_size, void* d_ws, size_t ws_size,
                              hipStream_t stream) {
  (void)in_sizes; (void)n_in; (void)out_size; (void)ws_size;
  const float* inputs   = (const float*)d_in[0];
  const float* maskin   = (const float*)d_in[1];
  const float* features = (const float*)d_in[2];
  const int*   labels   = (const int*)d_in[3];
  const int*   labels2  = (const int*)d_in[4];
  const int*   indexes  = (const int*)d_in[5];
  const int*   epoch    = (const int*)d_in[6];
  const int*   back     = (const int*)d_in[7];
  float* out = (float*)d_out;

  // workspace layout (all 4-byte elements)
  char* ws = (char*)d_ws;
  float* Fcls = (float*)ws;                              // NCP*NF
  float* sim  = Fcls + (size_t)NCP * NF;                 // BATCH*NCP
  float* nums = sim + (size_t)BATCH * NCP;               // NCP
  int*   tg   = (int*)(nums + NCP);                      // BATCH
  int*   tg2  = tg + BATCH;                              // BATCH
  float* pos_e = (float*)(tg2 + BATCH);                  // BATCH
  float* neg_e = pos_e + BATCH;                          // BATCH
  int*   n_pos = (int*)(neg_e + BATCH);                  // BATCH
  int*   n_neg = n_pos + BATCH;                          // BATCH
  float* focal_sum = (float*)(n_neg + BATCH);            // 1
  float* csum      = focal_sum + 1;                      // 1
  size_t totalFloats = (size_t)((char*)(csum + 1) - ws) / 4;

  k_zero<<<(unsigned)((totalFloats + 255) / 256), 256, 0, stream>>>((float*)ws,
                                                                    (int)totalFloats);
  k_gather<<<1, 256, 0, stream>>>(labels, labels2, indexes, tg, tg2);
  k_agg<<<NC / 8, 256, 0, stream>>>(features, labels, Fcls, nums);
  k_gemm_sim<<<dim3(NCP / 128, BATCH / 16), 256, 0, stream>>>(inputs, Fcls, nums, sim);
  k_focal<<<BATCH, 256, 0, stream>>>(sim, nums, tg, focal_sum);
  k_contras<<<BATCH, 256, 0, stream>>>(inputs, maskin, csum);
  k_gemm_ins<<<dim3(NS / 128, BATCH / 16), 256, 0, stream>>>(inputs, features, labels,
                                                             labels2, tg, tg2, pos_e,
                                                             neg_e, n_pos, n_neg);
  k_final<<<1, 256, 0, stream>>>(pos_e, neg_e, n_pos, n_neg, focal_sum, csum, epoch,
                                 back, out);
}